// RelationalKenn_59717225284038
// MI455X (gfx1250) — compile-verified
//
#include <hip/hip_runtime.h>
#include <stdint.h>

#define N_U     16
#define N_B     8
#define N_NODES 500000
#define N_EDGES 2000000
#define TPB     256

#if defined(__gfx1250__) && __has_builtin(__builtin_amdgcn_tensor_load_to_lds) && __has_builtin(__builtin_amdgcn_s_wait_tensorcnt)
#define USE_TDM 1
#else
#define USE_TDM 0
#endif

typedef __attribute__((ext_vector_type(4))) unsigned int u32x4;
typedef __attribute__((ext_vector_type(8))) int          i32x8;
typedef __attribute__((ext_vector_type(4))) int          i32x4;

__device__ __forceinline__ void softmax2(float v0, float v1, float& p0, float& p1) {
  float m  = fmaxf(v0, v1);
  float e0 = __expf(v0 - m), e1 = __expf(v1 - m);
  float inv = 1.0f / (e0 + e1);
  p0 = e0 * inv; p1 = e1 * inv;
}

__device__ __forceinline__ void softmax3(float v0, float v1, float v2,
                                         float& p0, float& p1, float& p2) {
  float m  = fmaxf(v0, fmaxf(v1, v2));
  float e0 = __expf(v0 - m), e1 = __expf(v1 - m), e2 = __expf(v2 - m);
  float inv = 1.0f / (e0 + e1 + e2);
  p0 = e0 * inv; p1 = e1 * inv; p2 = e2 * inv;
}

// ---------------------------------------------------------------------------
// Pass 1: per-node unary enhancement.
//   u = unary + sum_clauses w * softmax(signed literals) * signs
//   Clauses touch disjoint columns {0,1},{2,3},{4,5,6} -> in-place per group.
//   Also emits the compact 2-column gather table uc[n] = u[n][0:2] (4 MB,
//   L2-resident for the edge pass) and zero-fills the two scatter buffers.
// ---------------------------------------------------------------------------
__global__ __launch_bounds__(TPB) void kenn_node_kernel(
    const float* __restrict__ unary, const float* __restrict__ ucw,
    float* __restrict__ out_u, float* __restrict__ uc,
    float* __restrict__ s1, float* __restrict__ s2) {
  int n = blockIdx.x * TPB + threadIdx.x;
  if (n >= N_NODES) return;

  const float4* row = (const float4*)(unary + (size_t)n * N_U);
  float4 r0 = row[0], r1 = row[1], r2 = row[2], r3 = row[3];
  float w0 = ucw[0], w1 = ucw[1], w2 = ucw[2];

  { // clause 0: cols {0,1}, signs {-1,+1}
    float p0, p1; softmax2(-r0.x, r0.y, p0, p1);
    r0.x -= w0 * p0; r0.y += w0 * p1;
  }
  { // clause 1: cols {2,3}, signs {-1,+1}
    float p0, p1; softmax2(-r0.z, r0.w, p0, p1);
    r0.z -= w1 * p0; r0.w += w1 * p1;
  }
  { // clause 2: cols {4,5,6}, signs {-1,+1,+1}
    float p0, p1, p2; softmax3(-r1.x, r1.y, r1.z, p0, p1, p2);
    r1.x -= w2 * p0; r1.y += w2 * p1; r1.z += w2 * p2;
  }

  float4* orow = (float4*)(out_u + (size_t)n * N_U);
  orow[0] = r0; orow[1] = r1; orow[2] = r2; orow[3] = r3;

  ((float2*)uc)[n] = make_float2(r0.x, r0.y);
  ((float2*)s1)[n] = make_float2(0.f, 0.f);   // zero the ".set" scatter targets
  ((float2*)s2)[n] = make_float2(0.f, 0.f);
}

// ---------------------------------------------------------------------------
// Pass 2: per-edge binary enhancement.
//   Block handles 256 consecutive edges; their binary rows are one contiguous
//   8 KB slab -> streamed into LDS by the Tensor Data Mover while each lane
//   issues its two random 8-byte gathers from the compact uc table.
//   Binary clauses only touch joined cols {0,16,32} and {1,17,33}:
//     ux -> plain store to s1[i1]  (set semantics; dup index = one winner)
//     uy -> plain store to s2[i2]
//     db -> binary_out cols 0,1
// ---------------------------------------------------------------------------
__global__ __launch_bounds__(TPB) void kenn_edge_kernel(
    const float* __restrict__ binary, const int* __restrict__ eidx,
    const float* __restrict__ ew, const float* __restrict__ bcw,
    const float* __restrict__ uc,
    float* __restrict__ s1, float* __restrict__ s2,
    float* __restrict__ out_b) {
  __shared__ __align__(16) float smem[TPB * N_B];

  const int e0    = blockIdx.x * TPB;
  const int nrows = (N_EDGES - e0 < TPB) ? (N_EDGES - e0) : TPB;

#if USE_TDM
  // TDM ignores EXEC and issues once per executing wave: restrict to wave 0
  // (only wave 0 reaches this divergent region; waves 1..7 branch around it).
  if (threadIdx.x == 0) {
    uint64_t ga    = (uint64_t)(const void*)(binary + (size_t)e0 * N_B);
    uint32_t lds   = (uint32_t)(uintptr_t)(void*)&smem[0]; // flat LDS addr low 32 = LDS offset
    uint32_t nelem = (uint32_t)nrows * N_B;                // dwords, <= 2048

    u32x4 g0;
    g0[0] = 1u;                                            // count=1 (valid user D#)
    g0[1] = lds;                                           // lds_addr
    g0[2] = (uint32_t)ga;                                  // global_addr[31:0]
    g0[3] = ((uint32_t)(ga >> 32) & 0x01FFFFFFu)           // global_addr[56:32]
          | (2u << 30);                                    // type = 2 ("image")
    i32x8 g1;
    g1[0] = (int)(2u << 16);                               // data_size = 4 bytes
    g1[1] = (int)((nelem & 0xFFFFu) << 16);                // tensor_dim0[15:0]
    g1[2] = (int)((nelem >> 16) | (1u << 16));             // tensor_dim0[31:16] | tensor_dim1=1
    g1[3] = (int)(nelem << 16);                            // tile_dim0 (<=2048)
    g1[4] = 1;                                             // tile_dim1 = 1, tile_dim2 = 0
    g1[5] = (int)nelem;                                    // tensor_dim0_stride
    g1[6] = 0;
    g1[7] = 0;
    i32x4 gz4 = {0, 0, 0, 0};                              // groups 2/3 unused (<=2D)
    i32x8 gz8 = {0, 0, 0, 0, 0, 0, 0, 0};
    __builtin_amdgcn_tensor_load_to_lds(g0, g1, gz4, gz4, gz8, 0);
  }
#else
  if ((int)threadIdx.x < nrows) {
    float4* dst = (float4*)&smem[threadIdx.x * N_B];
    const float4* src = (const float4*)(binary + (size_t)(e0 + threadIdx.x) * N_B);
    dst[0] = src[0];
    dst[1] = src[1];
  }
#endif

  // Overlap: random gathers from the 4 MB (L2-resident) compact u table
  // while the TDM streams the binary slab into LDS.
  const int  e   = e0 + (int)threadIdx.x;
  const bool act = e < N_EDGES;
  int    i1 = 0, i2 = 0;
  float  we = 0.f;
  float2 ua = make_float2(0.f, 0.f), ub = make_float2(0.f, 0.f);
  if (act) {
    i1 = eidx[e];
    i2 = eidx[N_EDGES + e];
    we = ew[e];
    ua = ((const float2*)uc)[i1];
    ub = ((const float2*)uc)[i2];
  }
  const float wA = bcw[0], wB = bcw[1];

#if USE_TDM
  if (threadIdx.x == 0) __builtin_amdgcn_s_wait_tensorcnt(0);
#endif
  __syncthreads();

  if (act) {
    const float4* brow = (const float4*)&smem[threadIdx.x * N_B];
    float4 b0 = brow[0], b1 = brow[1];

    // clause A: joined cols {0,16,32} = {u1[0], u2[0], bin[0]}, signs {-1,-1,+1}
    float pA0, pA1, pA2; softmax3(-ua.x, -ub.x, b0.x, pA0, pA1, pA2);
    // clause B: joined cols {1,17,33} = {u1[1], u2[1], bin[1]}, signs {-1,+1,+1}
    float pB0, pB1, pB2; softmax3(-ua.y,  ub.y, b0.y, pB0, pB1, pB2);

    ((float2*)s1)[i1] = make_float2(-wA * pA0 * we, -wB * pB0 * we); // ux cols 0,1
    ((float2*)s2)[i2] = make_float2(-wA * pA1 * we,  wB * pB1 * we); // uy cols 0,1

    b0.x += wA * pA2 * we;                                           // db col 0
    b0.y += wB * pB2 * we;                                           // db col 1
    float4* orow = (float4*)(out_b + (size_t)e * N_B);
    orow[0] = b0; orow[1] = b1;
  }
}

// ---------------------------------------------------------------------------
// Pass 3: out_u[:, 0:2] += s1 + s2   (delta is zero in cols 2..15)
// ---------------------------------------------------------------------------
__global__ __launch_bounds__(TPB) void kenn_finish_kernel(
    float* __restrict__ out_u, const float* __restrict__ s1,
    const float* __restrict__ s2) {
  int n = blockIdx.x * TPB + threadIdx.x;
  if (n >= N_NODES) return;
  float2 d1 = ((const float2*)s1)[n];
  float2 d2 = ((const float2*)s2)[n];
  float2* p = (float2*)(out_u + (size_t)n * N_U);
  float2 v = *p;
  v.x += d1.x + d2.x;
  v.y += d1.y + d2.y;
  *p = v;
}

extern "C" void kernel_launch(void* const* d_in, const int* in_sizes, int n_in,
                              void* d_out, int out_size, void* d_ws, size_t ws_size,
                              hipStream_t stream) {
  (void)in_sizes; (void)n_in; (void)out_size; (void)ws_size;

  const float* unary  = (const float*)d_in[0];   // (N_NODES, 16) f32
  const float* binary = (const float*)d_in[1];   // (N_EDGES, 8)  f32
  const int*   eidx   = (const int*)  d_in[2];   // (2, N_EDGES)  i32
  const float* ew     = (const float*)d_in[3];   // (N_EDGES,)    f32
  const float* ucw    = (const float*)d_in[4];   // (3,) f32
  const float* bcw    = (const float*)d_in[5];   // (2,) f32

  float* out_u = (float*)d_out;                      // (N_NODES, 16)
  float* out_b = out_u + (size_t)N_NODES * N_U;      // (N_EDGES, 8)

  float* uc = (float*)d_ws;                          // (N_NODES, 2) compact u[:,0:2]
  float* s1 = uc + (size_t)N_NODES * 2;              // (N_NODES, 2) i1-scatter
  float* s2 = s1 + (size_t)N_NODES * 2;              // (N_NODES, 2) i2-scatter
                                                     // total ws: 12 MB

  const int nblk_n = (N_NODES + TPB - 1) / TPB;
  const int nblk_e = (N_EDGES + TPB - 1) / TPB;

  kenn_node_kernel<<<nblk_n, TPB, 0, stream>>>(unary, ucw, out_u, uc, s1, s2);
  kenn_edge_kernel<<<nblk_e, TPB, 0, stream>>>(binary, eidx, ew, bcw, uc, s1, s2, out_b);
  kenn_finish_kernel<<<nblk_n, TPB, 0, stream>>>(out_u, s1, s2);
}